// GraphConv_24154896073115
// MI455X (gfx1250) — compile-verified
//
#include <hip/hip_runtime.h>
#include <hip/hip_bf16.h>

typedef __attribute__((ext_vector_type(2))) float v2f;
typedef __attribute__((ext_vector_type(8))) float v8f;

#define D_FEAT 128

// ---------------------------------------------------------------------------
// K1: deg[v] = 1.0f  (self-loop contribution)
// ---------------------------------------------------------------------------
__global__ void k_init_deg(float* __restrict__ deg, int n_nodes) {
    int i = blockIdx.x * blockDim.x + threadIdx.x;
    if (i < n_nodes) deg[i] = 1.0f;
}

// ---------------------------------------------------------------------------
// K2: deg[dst[e]] += 1   (in-degree)
// ---------------------------------------------------------------------------
__global__ void k_count_deg(const int* __restrict__ dst, float* __restrict__ deg,
                            int n_edges) {
    int e = blockIdx.x * blockDim.x + threadIdx.x;
    if (e < n_edges) {
        __hip_atomic_fetch_add(&deg[dst[e]], 1.0f,
                               __ATOMIC_RELAXED, __HIP_MEMORY_SCOPE_AGENT);
    }
}

// ---------------------------------------------------------------------------
// K3: norm[v] = rsqrt(deg[v]);  acc[v][f] = x[v][f] * norm[v]
//     (acc initialized with the pre-normalized self-loop term)
// ---------------------------------------------------------------------------
__global__ void k_norm_init(const float* __restrict__ x,
                            const float* __restrict__ deg,
                            float* __restrict__ norm,
                            float* __restrict__ acc, int total) {
    int i = blockIdx.x * blockDim.x + threadIdx.x;
    if (i >= total) return;
    int v = i >> 7;                       // / D_FEAT
    float nv = rsqrtf(deg[v]);            // deg >= 1 always (self-loop)
    acc[i] = x[i] * nv;
    if ((i & (D_FEAT - 1)) == 0) norm[v] = nv;
}

// ---------------------------------------------------------------------------
// K4: edge scatter. One wave32 per edge: lane l handles features [4l, 4l+4).
//     acc[dst] += x[src] * norm[src]   via global_atomic_add_f32.
//     src/dst/norm loads are wave-uniform -> scalarized by compiler; the
//     float4 feature loads and the 4 atomics per lane are fully coalesced.
// ---------------------------------------------------------------------------
__global__ void __launch_bounds__(256) k_edge_scatter(
        const float* __restrict__ x, const int* __restrict__ src,
        const int* __restrict__ dst, const float* __restrict__ norm,
        float* __restrict__ acc, int n_edges) {
    int gtid = blockIdx.x * blockDim.x + threadIdx.x;
    int wave = gtid >> 5;
    int lane = threadIdx.x & 31;
    if (wave >= n_edges) return;
    int s = src[wave];
    int d = dst[wave];
    float ns = norm[s];
    const float4* xrow = (const float4*)(x + (size_t)s * D_FEAT);
    float4 v = xrow[lane];
    float* arow = acc + (size_t)d * D_FEAT + lane * 4;
    __hip_atomic_fetch_add(arow + 0, v.x * ns, __ATOMIC_RELAXED, __HIP_MEMORY_SCOPE_AGENT);
    __hip_atomic_fetch_add(arow + 1, v.y * ns, __ATOMIC_RELAXED, __HIP_MEMORY_SCOPE_AGENT);
    __hip_atomic_fetch_add(arow + 2, v.z * ns, __ATOMIC_RELAXED, __HIP_MEMORY_SCOPE_AGENT);
    __hip_atomic_fetch_add(arow + 3, v.w * ns, __ATOMIC_RELAXED, __HIP_MEMORY_SCOPE_AGENT);
}

// ---------------------------------------------------------------------------
// K5: out[m, :] = (acc[node] * norm[node]) @ W^T  for node = dst_nodes[m],
//     using V_WMMA_F32_16X16X4_F32 (full f32, wave32).
//     Block = 256 threads = 8 waves. Block -> 16 output rows (one M tile);
//     wave w -> output cols [16w, 16w+16). K=128 -> 32 chained WMMAs.
//     norm[node] is a per-row constant, so it is applied to the accumulator
//     at store time instead of scaling A inside the K loop.
//
//     f32 WMMA operand layout (16x16x4):
//       A (16x4): lane = {hi,lr}: lr=M row, VGPR0/1 = K = 2*hi + {0,1}
//       B (4x16): lane = {hi,lc}: lc=N col, VGPR0/1 = K = 2*hi + {0,1}
//       C/D (16x16): VGPR r: lanes 0-15 -> M=r, lanes 16-31 -> M=8+r; N=lane%16
// ---------------------------------------------------------------------------
__global__ void __launch_bounds__(256) k_wmma_gemm_gather(
        const float* __restrict__ acc, const float* __restrict__ norm,
        const float* __restrict__ W, const int* __restrict__ dst_nodes,
        float* __restrict__ out, int n_dst) {
    const int lane = threadIdx.x & 31;
    const int wv   = threadIdx.x >> 5;          // 0..7 : N tile
    const int lr   = lane & 15;                 // A row / B col within tile
    const int hi   = lane >> 4;                 // selects K pair {2hi, 2hi+1}
    const int tile_m = blockIdx.x;

    // A row pointer (gathered through dst_nodes; clamp for tail rows)
    int m = tile_m * 16 + lr;
    int mm = (m < n_dst) ? m : (n_dst - 1);
    const float* arow = acc + (size_t)dst_nodes[mm] * D_FEAT;
    // B: out col n = wv*16 + lr corresponds to W row n (out = h @ W^T)
    const float* brow = W + (size_t)(wv * 16 + lr) * D_FEAT;

    v8f c = {0.f, 0.f, 0.f, 0.f, 0.f, 0.f, 0.f, 0.f};
#pragma unroll
    for (int kk = 0; kk < D_FEAT; kk += 4) {
        float2 av = *(const float2*)(arow + kk + hi * 2);
        float2 bv = *(const float2*)(brow + kk + hi * 2);
        v2f a; a.x = av.x; a.y = av.y;
        v2f b; b.x = bv.x; b.y = bv.y;
        // 8 args: (neg_a, A, neg_b, B, c_mod, C, reuse_a, reuse_b)
        c = __builtin_amdgcn_wmma_f32_16x16x4_f32(
                false, a, false, b, (short)0, c, false, false);
    }

    // Store D, applying the post-norm scale of each output row.
    const int ccol  = lane & 15;
    const int crow0 = (lane >> 4) * 8;
    const int n0    = wv * 16;
#pragma unroll
    for (int r = 0; r < 8; ++r) {
        int mr = tile_m * 16 + crow0 + r;
        if (mr < n_dst) {
            float s = norm[dst_nodes[mr]];
            out[(size_t)mr * D_FEAT + n0 + ccol] = c[r] * s;
        }
    }
}

// ---------------------------------------------------------------------------
// Launcher
// Inputs (setup_inputs order): 0:x [n,128] f32, 1:W [128,128] f32,
//                              2:src [E] i32, 3:dst [E] i32, 4:dst_nodes [M] i32
// Output: out [M,128] f32
// Workspace: deg[n] | norm[n] | acc[n*128]   (floats)
// ---------------------------------------------------------------------------
extern "C" void kernel_launch(void* const* d_in, const int* in_sizes, int n_in,
                              void* d_out, int out_size, void* d_ws, size_t ws_size,
                              hipStream_t stream) {
    const float* x         = (const float*)d_in[0];
    const float* W         = (const float*)d_in[1];
    const int*   src       = (const int*)d_in[2];
    const int*   dst       = (const int*)d_in[3];
    const int*   dst_nodes = (const int*)d_in[4];
    float*       out       = (float*)d_out;

    const int n_nodes = in_sizes[0] / D_FEAT;
    const int n_edges = in_sizes[2];
    const int n_dst   = in_sizes[4];
    const int total   = n_nodes * D_FEAT;

    float* deg  = (float*)d_ws;
    float* norm = deg + n_nodes;
    float* acc  = norm + n_nodes;

    // K1: deg = 1 (self-loops)
    k_init_deg<<<(n_nodes + 255) / 256, 256, 0, stream>>>(deg, n_nodes);
    // K2: in-degree
    k_count_deg<<<(n_edges + 255) / 256, 256, 0, stream>>>(dst, deg, n_edges);
    // K3: norm = rsqrt(deg); acc = x * norm (self-loop term)
    k_norm_init<<<(total + 255) / 256, 256, 0, stream>>>(x, deg, norm, acc, total);
    // K4: edge scatter (one wave32 per edge, 8 waves per block)
    {
        long long nthreads = (long long)n_edges * 32;
        int blocks = (int)((nthreads + 255) / 256);
        k_edge_scatter<<<blocks, 256, 0, stream>>>(x, src, dst, norm, acc, n_edges);
    }
    // K5: WMMA f32 GEMM with fused post-norm + dst_nodes gather
    k_wmma_gemm_gather<<<(n_dst + 15) / 16, 256, 0, stream>>>(
        acc, norm, W, dst_nodes, out, n_dst);
}